// TwoBodyEquiGraphConv_31499290149199
// MI455X (gfx1250) — compile-verified
//
#include <hip/hip_runtime.h>
#include <math.h>

#define NN 50000
#define EE 400000
#define FD 64
#define STLD 68            // 16x64 stage tile with +4 pad (bank-conflict free)
#define TILE_F (16 * STLD) // floats per stage tile

typedef __attribute__((ext_vector_type(2))) float v2f;
typedef __attribute__((ext_vector_type(8))) float v8f;

// D = A(16x4 f32) * B(4x16 f32) + C, exact fp32 on the matrix pipe
__device__ __forceinline__ v8f wmma4(v2f a, v2f b, v8f c) {
  return __builtin_amdgcn_wmma_f32_16x16x4_f32(false, a, false, b, (short)0, c,
                                               false, false);
}

__device__ __forceinline__ float sigm(float x) { return 1.0f / (1.0f + expf(-x)); }

__device__ __forceinline__ void atomic_add_f32(float* p, float v) {
  __hip_atomic_fetch_add(p, v, __ATOMIC_RELAXED, __HIP_MEMORY_SCOPE_AGENT);
}

// Wave-local LDS RAW fence: LDS ops from one wave issue in order, so waiting
// DScnt==0 makes prior wave-private stage writes visible to all lanes.
__device__ __forceinline__ void lds_fence() {
  asm volatile("s_wait_dscnt 0x0" ::: "memory");
}

// Copy W[K][N] (row-major) into pair-transposed LDS layout:
//   dst_pair[n*PSTR + p] = {W[2p][n], W[2p+1][n]},  PSTR = K/2 + 2
// -> one ds_load_b64 per B fragment, conflict-free banking.
__device__ __forceinline__ void stage_weight(float* dst, const float* W,
                                             int K, int N, int tid, int nthr) {
  const int PSTR = (K >> 1) + 2;
  for (int i = tid; i < K * N; i += nthr) {
    int k = i / N;
    int n = i - k * N;
    dst[2 * (n * PSTR + (k >> 1)) + (k & 1)] = W[i];
  }
}

__device__ __forceinline__ v2f bfrag(const float* Wl, int PSTR, int col,
                                     int k0, int half) {
  return *(const v2f*)&Wl[2 * (col * PSTR + (k0 >> 1) + half)];
}

extern "C" __global__ void zero_kernel(float* __restrict__ p, long n) {
  long i = (long)blockIdx.x * blockDim.x + threadIdx.x;
  if (i < n) p[i] = 0.0f;
}

// ---------------- EDGE KERNEL: 16 waves/block, one wave per 16-edge tile ----
extern "C" __global__ void __launch_bounds__(512)
edge_kernel(const float* __restrict__ node_s, const float* __restrict__ node_v,
            const float* __restrict__ edge_s, const float* __restrict__ edge_v,
            const float* __restrict__ dist,   const float* __restrict__ vctr,
            const int*   __restrict__ src,    const int*   __restrict__ dst,
            const float* __restrict__ W_nn,   const float* __restrict__ b_nn,
            const float* __restrict__ W_ep,   const float* __restrict__ b_ep,
            const float* __restrict__ Wg1,    const float* __restrict__ bg1,
            const float* __restrict__ Wg2,    const float* __restrict__ bg2,
            const float* __restrict__ W_ev,   const float* __restrict__ b_ev,
            float* __restrict__ es_out, float* __restrict__ ev_out,
            float* __restrict__ nes, float* __restrict__ nev,
            float* __restrict__ deg)
{
  extern __shared__ float sm[];
  float* Wnn_l = sm;                  // 64 cols * 66 pairs * 2 = 8448
  float* Wep_l = Wnn_l + 8448;        // 64 * 34 * 2 = 4352
  float* Wg1_l = Wep_l + 4352;        // 4352
  float* Wg2_l = Wg1_l + 4352;        // 128 * 34 * 2 = 8704
  float* Wev_l = Wg2_l + 8704;        // 192 * 34 * 2 = 13056
  float* stg0  = Wev_l + 13056;       // 16 waves * TILE_F

  const int tid = threadIdx.x;
  stage_weight(Wnn_l, W_nn, 128, 64, tid, 512);
  stage_weight(Wep_l, W_ep, 64, 64, tid, 512);
  stage_weight(Wg1_l, Wg1, 64, 64, tid, 512);
  stage_weight(Wg2_l, Wg2, 64, 128, tid, 512);
  stage_weight(Wev_l, W_ev, 64, 192, tid, 512);
  __syncthreads();

  const int lane = tid & 31;
  const int half = lane >> 4;
  const int sub  = lane & 15;
  float* stage = stg0 + (tid >> 5) * TILE_F;

  // loop-invariant biases
  float Bnn[4], Bep[4], Bg1[4], Bg2v[4], Bg2g[4], Bev[12];
#pragma unroll
  for (int t = 0; t < 4; t++) {
    Bnn[t]  = b_nn[t * 16 + sub];
    Bep[t]  = b_ep[t * 16 + sub];
    Bg1[t]  = bg1[t * 16 + sub];
    Bg2v[t] = bg2[t * 16 + sub];
    Bg2g[t] = bg2[64 + t * 16 + sub];
    Bev[t]     = b_ev[t * 16 + sub];
    Bev[4 + t] = b_ev[64 + t * 16 + sub];
    Bev[8 + t] = b_ev[128 + t * 16 + sub];
  }

  const int wave0 = blockIdx.x * 16 + (tid >> 5);
  const int nwv   = gridDim.x * 16;

#pragma unroll 1
  for (int tile = wave0; tile < EE / 16; tile += nwv) {
    const int e0 = tile << 4;

    // per-edge metadata held by lanes 0..15 (edge m = lane)
    float c_l = 0.f, vx = 0.f, vy = 0.f, vz = 0.f;
    int sv = 0, dv = 0;
    if (lane < 16) {
      int e = e0 + lane;
      float dd = dist[e];
      c_l = (dd < 5.0f) ? 0.5f * (cosf(0.6283185307179587f * dd) + 1.0f) : 0.0f;
      sv = src[e];
      dv = dst[e];
      vx = vctr[e * 3];
      vy = vctr[e * 3 + 1];
      vz = vctr[e * 3 + 2];
      atomic_add_f32(&deg[dv], 1.0f);
    }
    const int isrc = __shfl(sv, sub, 32);
    const int idst = __shfl(dv, sub, 32);
    float cR[8];
    int dR[8];
#pragma unroll
    for (int r = 0; r < 8; r++) {
      int m = r + 8 * half;
      cR[r] = __shfl(c_l, m, 32);
      dR[r] = __shfl(dv, m, 32);
    }

    // ---- GEMM1: nn = [ns_src|ns_dst]@W_nn ; GEMM2: ep = edge_s@W_ep ; msg
    {
      const float* rs = node_s + (size_t)isrc * FD + 2 * half;
      const float* rd = node_s + (size_t)idst * FD + 2 * half;
      const float* re = edge_s + (size_t)(e0 + sub) * FD + 2 * half;
      v8f nn[4] = {};
      v8f ep[4] = {};
#pragma unroll
      for (int k0 = 0; k0 < 128; k0 += 4) {
        v2f a = (k0 < 64) ? *(const v2f*)(rs + k0) : *(const v2f*)(rd + k0 - 64);
        v2f b0 = bfrag(Wnn_l, 66, sub, k0, half);
        v2f b1 = bfrag(Wnn_l, 66, 16 + sub, k0, half);
        v2f b2 = bfrag(Wnn_l, 66, 32 + sub, k0, half);
        v2f b3 = bfrag(Wnn_l, 66, 48 + sub, k0, half);
        nn[0] = wmma4(a, b0, nn[0]);
        nn[1] = wmma4(a, b1, nn[1]);
        nn[2] = wmma4(a, b2, nn[2]);
        nn[3] = wmma4(a, b3, nn[3]);
      }
#pragma unroll
      for (int k0 = 0; k0 < 64; k0 += 4) {
        v2f a = *(const v2f*)(re + k0);
        v2f b0 = bfrag(Wep_l, 34, sub, k0, half);
        v2f b1 = bfrag(Wep_l, 34, 16 + sub, k0, half);
        v2f b2 = bfrag(Wep_l, 34, 32 + sub, k0, half);
        v2f b3 = bfrag(Wep_l, 34, 48 + sub, k0, half);
        ep[0] = wmma4(a, b0, ep[0]);
        ep[1] = wmma4(a, b1, ep[1]);
        ep[2] = wmma4(a, b2, ep[2]);
        ep[3] = wmma4(a, b3, ep[3]);
      }
#pragma unroll
      for (int t = 0; t < 4; t++)
#pragma unroll
        for (int r = 0; r < 8; r++)
          stage[(r + 8 * half) * STLD + t * 16 + sub] =
              (nn[t][r] + Bnn[t]) * (ep[t][r] + Bep[t]);
    }
    lds_fence();

    // ---- GEMM3: h = silu(msg @ Wg1 + bg1) -> stage
    {
      v8f h[4] = {};
#pragma unroll
      for (int k0 = 0; k0 < 64; k0 += 4) {
        v2f a = *(const v2f*)&stage[sub * STLD + k0 + 2 * half];
        v2f b0 = bfrag(Wg1_l, 34, sub, k0, half);
        v2f b1 = bfrag(Wg1_l, 34, 16 + sub, k0, half);
        v2f b2 = bfrag(Wg1_l, 34, 32 + sub, k0, half);
        v2f b3 = bfrag(Wg1_l, 34, 48 + sub, k0, half);
        h[0] = wmma4(a, b0, h[0]);
        h[1] = wmma4(a, b1, h[1]);
        h[2] = wmma4(a, b2, h[2]);
        h[3] = wmma4(a, b3, h[3]);
      }
#pragma unroll
      for (int t = 0; t < 4; t++)
#pragma unroll
        for (int r = 0; r < 8; r++) {
          float x = h[t][r] + Bg1[t];
          stage[(r + 8 * half) * STLD + t * 16 + sub] = x * sigm(x);
        }
    }
    lds_fence();

    // ---- GEMM4: hv = h @ Wg2 (Nout=128); e_s_upd = val*sigmoid(gate)*c
    {
      v8f hv[8] = {};
#pragma unroll
      for (int k0 = 0; k0 < 64; k0 += 4) {
        v2f a = *(const v2f*)&stage[sub * STLD + k0 + 2 * half];
#pragma unroll
        for (int t = 0; t < 8; t++) {
          v2f b = bfrag(Wg2_l, 34, t * 16 + sub, k0, half);
          hv[t] = wmma4(a, b, hv[t]);
        }
      }
#pragma unroll
      for (int t = 0; t < 4; t++) {
#pragma unroll
        for (int r = 0; r < 8; r++) {
          int m = r + 8 * half;
          int col = t * 16 + sub;
          float val = hv[t][r] + Bg2v[t];
          float gt  = hv[4 + t][r] + Bg2g[t];
          float esu = val * sigm(gt) * cR[r];
          size_t eidx = (size_t)(e0 + m) * FD + col;
          es_out[eidx] = esu + edge_s[eidx];
          atomic_add_f32(&nes[(size_t)dR[r] * FD + col], esu);
          stage[m * STLD + col] = esu;
        }
      }
    }
    lds_fence();

    // broadcasts needed only by the vector stage
    int sR[8];
    float vxR[8], vyR[8], vzR[8];
#pragma unroll
    for (int r = 0; r < 8; r++) {
      int m = r + 8 * half;
      sR[r]  = __shfl(sv, m, 32);
      vxR[r] = __shfl(vx, m, 32);
      vyR[r] = __shfl(vy, m, 32);
      vzR[r] = __shfl(vz, m, 32);
    }

    // ---- GEMM5: ch = e_s_upd @ W_ev (Nout=192); edge vector update
#pragma unroll 1
    for (int j = 0; j < 4; j++) {
      v8f cn = {}, ce = {}, cr = {};
#pragma unroll
      for (int k0 = 0; k0 < 64; k0 += 4) {
        v2f a = *(const v2f*)&stage[sub * STLD + k0 + 2 * half];
        v2f b0 = bfrag(Wev_l, 34, j * 16 + sub, k0, half);
        v2f b1 = bfrag(Wev_l, 34, 64 + j * 16 + sub, k0, half);
        v2f b2 = bfrag(Wev_l, 34, 128 + j * 16 + sub, k0, half);
        cn = wmma4(a, b0, cn);
        ce = wmma4(a, b1, ce);
        cr = wmma4(a, b2, cr);
      }
#pragma unroll
      for (int r = 0; r < 8; r++) {
        int m = r + 8 * half;
        int col = j * 16 + sub;
        float nch = cn[r] + Bev[j];
        float ech = ce[r] + Bev[4 + j];
        float rch = cr[r] + Bev[8 + j];
        float c = cR[r];
        size_t nvb = (size_t)sR[r] * 192 + col;
        size_t evb = (size_t)(e0 + m) * 192 + col;
        size_t agb = (size_t)dR[r] * 192 + col;
        float vR3[3] = {vxR[r], vyR[r], vzR[r]};
#pragma unroll
        for (int d = 0; d < 3; d++) {
          float evv = edge_v[evb + d * 64];
          float evu = (node_v[nvb + d * 64] * nch + evv * ech + vR3[d] * rch) * c;
          ev_out[evb + d * 64] = evu + evv;
          atomic_add_f32(&nev[agb + d * 64], evu);
        }
      }
    }
  }
}

// ---------------- NODE KERNEL: 8 waves/block, one wave per 16-node tile ----
extern "C" __global__ void __launch_bounds__(256)
node_kernel(const float* __restrict__ node_s, const float* __restrict__ node_v,
            const float* __restrict__ W_nvout, const float* __restrict__ W_nvch,
            const float* __restrict__ b_nvch,  const float* __restrict__ W_nvproj,
            const float* __restrict__ W_nsproj,const float* __restrict__ b_nsproj,
            const float* __restrict__ ln_g,    const float* __restrict__ ln_b,
            const float* __restrict__ cn_scale,
            const float* __restrict__ nes, const float* __restrict__ nev,
            const float* __restrict__ deg,
            float* __restrict__ ns_out, float* __restrict__ nv_out)
{
  extern __shared__ float sm[];
  float* Wvo_l = sm;                  // 192 * 34 * 2 = 13056
  float* Wvc_l = Wvo_l + 13056;       // 64 * 66 * 2 = 8448  (K=128)
  float* Wvp_l = Wvc_l + 8448;        // 128 * 34 * 2 = 8704
  float* Wsp_l = Wvp_l + 8704;        // 8704
  float* stg0  = Wsp_l + 8704;        // 8 waves * 4 * TILE_F

  const int tid = threadIdx.x;
  stage_weight(Wvo_l, W_nvout, 64, 192, tid, 256);
  stage_weight(Wvc_l, W_nvch, 128, 64, tid, 256);
  stage_weight(Wvp_l, W_nvproj, 64, 128, tid, 256);
  stage_weight(Wsp_l, W_nsproj, 64, 128, tid, 256);
  __syncthreads();

  const int lane = tid & 31;
  const int half = lane >> 4;
  const int sub  = lane & 15;
  float* wbase = stg0 + (tid >> 5) * (4 * TILE_F);
  float* o3s  = wbase;
  float* nvs0 = wbase + TILE_F;
  float* nvs1 = wbase + 2 * TILE_F;
  float* nvs2 = wbase + 3 * TILE_F;
  float* const nvsP[3] = {nvs0, nvs1, nvs2};

  // loop-invariant params
  float Bvc[4], Bs1[4], Bs2[4], Lg[4], Lb[4], Cs[4];
#pragma unroll
  for (int t = 0; t < 4; t++) {
    Bvc[t] = b_nvch[t * 16 + sub];
    Bs1[t] = b_nsproj[t * 16 + sub];
    Bs2[t] = b_nsproj[64 + t * 16 + sub];
    Lg[t]  = ln_g[t * 16 + sub];
    Lb[t]  = ln_b[t * 16 + sub];
    Cs[t]  = cn_scale[t * 16 + sub];
  }

  const int wave0 = blockIdx.x * 8 + (tid >> 5);
  const int nwv   = gridDim.x * 8;

#pragma unroll 1
  for (int tile = wave0; tile < NN / 16; tile += nwv) {
    const int n0 = tile << 4;

    float dn_l = 0.f;
    if (lane < 16) dn_l = 1.0f / fmaxf(deg[n0 + lane], 1.0f);
    const float dns = __shfl(dn_l, sub, 32);

    const float* aes = nes + (size_t)(n0 + sub) * FD + 2 * half;
    const float* aev = nev + (size_t)(n0 + sub) * 192 + 2 * half;
    const float* ao3 = &o3s[sub * STLD + 2 * half];

    // ---- o3 = n_ev @ W_nvout[:,128:192]; o3n -> o3s
#pragma unroll 1
    for (int j = 0; j < 4; j++) {
      v8f ss = {};
#pragma unroll
      for (int d = 0; d < 3; d++) {
        v8f o3 = {};
#pragma unroll
        for (int k0 = 0; k0 < 64; k0 += 4) {
          v2f a = *(const v2f*)(aev + d * 64 + k0);
          a.x *= dns;
          a.y *= dns;
          v2f b = bfrag(Wvo_l, 34, 128 + j * 16 + sub, k0, half);
          o3 = wmma4(a, b, o3);
        }
#pragma unroll
        for (int r = 0; r < 8; r++) ss[r] += o3[r] * o3[r];
      }
#pragma unroll
      for (int r = 0; r < 8; r++)
        o3s[(r + 8 * half) * STLD + j * 16 + sub] = sqrtf(ss[r]);
    }
    lds_fence();

    // ---- vch = [n_es | o3n] @ W_nvch + b (K=128)
    v8f vch[4] = {};
#pragma unroll
    for (int k0 = 0; k0 < 128; k0 += 4) {
      v2f a;
      if (k0 < 64) {
        a = *(const v2f*)(aes + k0);
        a.x *= dns;
        a.y *= dns;
      } else {
        a = *(const v2f*)(ao3 + (k0 - 64));
      }
      v2f b0 = bfrag(Wvc_l, 66, sub, k0, half);
      v2f b1 = bfrag(Wvc_l, 66, 16 + sub, k0, half);
      v2f b2 = bfrag(Wvc_l, 66, 32 + sub, k0, half);
      v2f b3 = bfrag(Wvc_l, 66, 48 + sub, k0, half);
      vch[0] = wmma4(a, b0, vch[0]);
      vch[1] = wmma4(a, b1, vch[1]);
      vch[2] = wmma4(a, b2, vch[2]);
      vch[3] = wmma4(a, b3, vch[3]);
    }
#pragma unroll
    for (int t = 0; t < 4; t++)
#pragma unroll
      for (int r = 0; r < 8; r++) vch[t][r] += Bvc[t];

    // ---- n_v_upd = o1*vch + o2; residual + CoorsNorm; stage n_v_upd
#pragma unroll 1
    for (int j = 0; j < 4; j++) {
      v8f nv3[3];
#pragma unroll
      for (int d = 0; d < 3; d++) {
        v8f o1 = {}, o2 = {};
#pragma unroll
        for (int k0 = 0; k0 < 64; k0 += 4) {
          v2f a = *(const v2f*)(aev + d * 64 + k0);
          a.x *= dns;
          a.y *= dns;
          v2f b0 = bfrag(Wvo_l, 34, j * 16 + sub, k0, half);
          v2f b1 = bfrag(Wvo_l, 34, 64 + j * 16 + sub, k0, half);
          o1 = wmma4(a, b0, o1);
          o2 = wmma4(a, b1, o2);
        }
#pragma unroll
        for (int r = 0; r < 8; r++) {
          float u = o1[r] * vch[j][r] + o2[r];
          nvsP[d][(r + 8 * half) * STLD + j * 16 + sub] = u;
          nv3[d][r] =
              u + node_v[(size_t)(n0 + r + 8 * half) * 192 + d * 64 + j * 16 + sub];
        }
      }
#pragma unroll
      for (int r = 0; r < 8; r++) {
        float nrm = sqrtf(nv3[0][r] * nv3[0][r] + nv3[1][r] * nv3[1][r] +
                          nv3[2][r] * nv3[2][r]);
        float inv = Cs[j] / fmaxf(nrm, 1e-8f);
        size_t base = (size_t)(n0 + r + 8 * half) * 192 + j * 16 + sub;
        nv_out[base]       = nv3[0][r] * inv;
        nv_out[base + 64]  = nv3[1][r] * inv;
        nv_out[base + 128] = nv3[2][r] * inv;
      }
    }
    lds_fence();

    // ---- p1/p2 on n_v_upd, s1/s2 on n_es; n_s_upd + residual
    v8f nst[4];
#pragma unroll
    for (int t = 0; t < 4; t++) {
      v8f accP = {};
#pragma unroll
      for (int d = 0; d < 3; d++) {
        v8f p1 = {}, p2 = {};
        const float* av = &nvsP[d][sub * STLD + 2 * half];
#pragma unroll
        for (int k0 = 0; k0 < 64; k0 += 4) {
          v2f a = *(const v2f*)(av + k0);
          v2f b0 = bfrag(Wvp_l, 34, t * 16 + sub, k0, half);
          v2f b1 = bfrag(Wvp_l, 34, 64 + t * 16 + sub, k0, half);
          p1 = wmma4(a, b0, p1);
          p2 = wmma4(a, b1, p2);
        }
#pragma unroll
        for (int r = 0; r < 8; r++) accP[r] += p1[r] * p2[r];
      }
      v8f s1 = {}, s2 = {};
#pragma unroll
      for (int k0 = 0; k0 < 64; k0 += 4) {
        v2f a = *(const v2f*)(aes + k0);
        a.x *= dns;
        a.y *= dns;
        v2f b0 = bfrag(Wsp_l, 34, t * 16 + sub, k0, half);
        v2f b1 = bfrag(Wsp_l, 34, 64 + t * 16 + sub, k0, half);
        s1 = wmma4(a, b0, s1);
        s2 = wmma4(a, b1, s2);
      }
#pragma unroll
      for (int r = 0; r < 8; r++) {
        float x1 = s1[r] + Bs1[t], x2 = s2[r] + Bs2[t];
        x1 = x1 * sigm(x1);
        x2 = x2 * sigm(x2);
        nst[t][r] = accP[r] * x1 + x2 +
                    node_s[(size_t)(n0 + r + 8 * half) * FD + t * 16 + sub];
      }
    }

    // ---- LayerNorm over F (row lives across the 16-lane half + 4 tiles)
#pragma unroll
    for (int r = 0; r < 8; r++) {
      float x0 = nst[0][r], x1 = nst[1][r], x2 = nst[2][r], x3 = nst[3][r];
      float s = x0 + x1 + x2 + x3;
      float q = x0 * x0 + x1 * x1 + x2 * x2 + x3 * x3;
#pragma unroll
      for (int off = 1; off < 16; off <<= 1) {
        s += __shfl_xor(s, off, 32);
        q += __shfl_xor(q, off, 32);
      }
      float mu  = s * (1.0f / 64.0f);
      float var = q * (1.0f / 64.0f) - mu * mu;
      float inv = rsqrtf(var + 1e-5f);
      size_t base = (size_t)(n0 + r + 8 * half) * FD + sub;
      ns_out[base]      = (x0 - mu) * inv * Lg[0] + Lb[0];
      ns_out[base + 16] = (x1 - mu) * inv * Lg[1] + Lb[1];
      ns_out[base + 32] = (x2 - mu) * inv * Lg[2] + Lb[2];
      ns_out[base + 48] = (x3 - mu) * inv * Lg[3] + Lb[3];
    }
  }
}

extern "C" void kernel_launch(void* const* d_in, const int* in_sizes, int n_in,
                              void* d_out, int out_size, void* d_ws, size_t ws_size,
                              hipStream_t stream) {
  const float* node_s = (const float*)d_in[0];
  const float* node_v = (const float*)d_in[1];
  const float* edge_s = (const float*)d_in[2];
  const float* edge_v = (const float*)d_in[3];
  const float* dist   = (const float*)d_in[4];
  const float* vctr   = (const float*)d_in[5];
  const int*   src    = (const int*)d_in[6];
  const int*   dst    = (const int*)d_in[7];
  const float* W_nn = (const float*)d_in[8];   const float* b_nn = (const float*)d_in[9];
  const float* W_ep = (const float*)d_in[10];  const float* b_ep = (const float*)d_in[11];
  const float* Wg1  = (const float*)d_in[12];  const float* bg1  = (const float*)d_in[13];
  const float* Wg2  = (const float*)d_in[14];  const float* bg2  = (const float*)d_in[15];
  const float* W_ev = (const float*)d_in[16];  const float* b_ev = (const float*)d_in[17];
  const float* W_nvout  = (const float*)d_in[18];
  const float* W_nvch   = (const float*)d_in[19];
  const float* b_nvch   = (const float*)d_in[20];
  const float* W_nvproj = (const float*)d_in[21];
  const float* W_nsproj = (const float*)d_in[22];
  const float* b_nsproj = (const float*)d_in[23];
  const float* ln_g     = (const float*)d_in[24];
  const float* ln_b     = (const float*)d_in[25];
  const float* cn_scale = (const float*)d_in[26];

  float* ws  = (float*)d_ws;
  float* nes = ws;                     // [N,64]
  float* nev = ws + (size_t)NN * FD;   // [N,3,64]
  float* deg = nev + (size_t)NN * 192; // [N]
  long tot = (long)NN * (FD + 192 + 1);
  zero_kernel<<<(unsigned)((tot + 255) / 256), 256, 0, stream>>>(ws, tot);

  float* out    = (float*)d_out;
  float* ns_out = out;
  float* nv_out = out + (size_t)NN * FD;
  float* es_out = nv_out + (size_t)NN * 192;
  float* ev_out = es_out + (size_t)EE * FD;

  // edge: 155.6KB weights + 16 * 4.25KB stage = 220KB dynamic LDS
  size_t edge_lds = (size_t)(8448 + 4352 + 4352 + 8704 + 13056 + 16 * TILE_F) * 4;
  edge_kernel<<<512, 512, edge_lds, stream>>>(
      node_s, node_v, edge_s, edge_v, dist, vctr, src, dst,
      W_nn, b_nn, W_ep, b_ep, Wg1, bg1, Wg2, bg2, W_ev, b_ev,
      es_out, ev_out, nes, nev, deg);

  // node: 155.6KB weights + 8 * 17KB stage = 288KB dynamic LDS
  size_t node_lds = (size_t)(13056 + 8448 + 8704 + 8704 + 8 * 4 * TILE_F) * 4;
  node_kernel<<<256, 256, node_lds, stream>>>(
      node_s, node_v, W_nvout, W_nvch, b_nvch, W_nvproj, W_nsproj, b_nsproj,
      ln_g, ln_b, cn_scale, nes, nev, deg, ns_out, nv_out);
}